// NNUE_42159398978365
// MI455X (gfx1250) — compile-verified
//
#include <hip/hip_runtime.h>

// ---------------------------------------------------------------------------
// NNUE forward on gfx1250 (MI455X, wave32)
//   h   = sum over bag of emb_table rows            [16384,128]  (d_ws)
//   pol = h @ policy_w.T + policy_b                 [16384,225]  (d_out[0:...])
//   val = tanh(MLP(h))                              [16384,1]    (d_out tail)
// Policy GEMM uses V_WMMA_F32_16X16X4_F32 (full f32 precision).
// ---------------------------------------------------------------------------

typedef __attribute__((ext_vector_type(2))) float v2f;
typedef __attribute__((ext_vector_type(8))) float v8f;

#define NNUE_BATCH 16384
#define NNUE_BAG   64
#define NNUE_L1    128
#define NNUE_POL   225

// ---------------------------------------------------------------------------
// Kernel A: EmbeddingBag(sum). One wave per batch row; lane l accumulates
// columns [4l, 4l+4). 64 coalesced 512B row reads per batch row; the 51.8 MB
// table is L2-resident on MI455X (192 MB L2), so this runs at L2 bandwidth.
// ---------------------------------------------------------------------------
__global__ void nnue_embed_bag(const int* __restrict__ indices,
                               const float* __restrict__ emb,
                               float* __restrict__ h) {
  const int wave = threadIdx.x >> 5;
  const int lane = threadIdx.x & 31;
  const int b = blockIdx.x * 8 + wave;

  const int* bag = indices + (long long)b * NNUE_BAG;

  float4 acc = make_float4(0.f, 0.f, 0.f, 0.f);
  #pragma unroll 4
  for (int j = 0; j < NNUE_BAG; ++j) {
    // prefetch a few bag entries ahead into the cache hierarchy
    if (j + 4 < NNUE_BAG) {
      __builtin_prefetch(emb + (long long)bag[j + 4] * NNUE_L1, 0, 0);
    }
    const long long idx = bag[j];
    const float4 v = ((const float4*)(emb + idx * NNUE_L1))[lane];
    acc.x += v.x; acc.y += v.y; acc.z += v.z; acc.w += v.w;
  }
  ((float4*)(h + (long long)b * NNUE_L1))[lane] = acc;
}

// ---------------------------------------------------------------------------
// Kernel B: policy GEMM via V_WMMA_F32_16X16X4_F32.
// One wave (32 threads) computes a 16x16 tile of pol.
//   A (16x4, MxK) layout: lanes 0-15 -> M=lane, VGPR0=K, VGPR1=K+1;
//                         lanes 16-31 -> M=lane-16, VGPR0=K+2, VGPR1=K+3.
//   B (4x16, KxN) layout mirrors A with N on lanes.
//   C/D (16x16): VGPR r -> M=r (lanes 0-15) / M=r+8 (lanes 16-31), N=lane%16.
// EXEC stays all-1s through the WMMAs (ISA requirement); only the final
// store is predicated for the 225-column tail.
// ---------------------------------------------------------------------------
__global__ void nnue_policy_wmma(const float* __restrict__ h,
                                 const float* __restrict__ pw,   // [225,128]
                                 const float* __restrict__ pb,   // [225]
                                 float* __restrict__ pol) {      // [16384,225]
  const int lane  = threadIdx.x;       // 0..31
  const int row0  = blockIdx.x * 16;   // batch tile
  const int col0  = blockIdx.y * 16;   // policy-col tile
  const int half  = lane >> 4;         // 0: K,K+1   1: K+2,K+3
  const int l15   = lane & 15;
  const int col   = col0 + l15;
  const int colc  = (col < NNUE_POL) ? col : (NNUE_POL - 1);  // clamp, no branch
  const int koff  = half * 2;

  const float* hrow = h  + (long long)(row0 + l15) * NNUE_L1 + koff;
  const float* wrow = pw + (long long)colc       * NNUE_L1 + koff;

  const float bias = pb[colc];
  v8f c;
  #pragma unroll
  for (int r = 0; r < 8; ++r) c[r] = bias;

  #pragma unroll
  for (int k0 = 0; k0 < NNUE_L1; k0 += 4) {
    const v2f a  = *(const v2f*)(hrow + k0);   // h[row][k0+koff .. +1]
    const v2f bm = *(const v2f*)(wrow + k0);   // pw[col][k0+koff .. +1] == B[k][col]
    c = __builtin_amdgcn_wmma_f32_16x16x4_f32(
        /*neg_a=*/false, a, /*neg_b=*/false, bm,
        /*c_mod=*/(short)0, c, /*reuse_a=*/false, /*reuse_b=*/false);
  }

  if (col < NNUE_POL) {
    const int rbase = row0 + half * 8;
    #pragma unroll
    for (int r = 0; r < 8; ++r) {
      pol[(long long)(rbase + r) * NNUE_POL + col] = c[r];
    }
  }
}

// ---------------------------------------------------------------------------
// Kernel C: fused value head. One wave per batch row; lane j owns output j
// of layers 1 and 2 (32 wide). Layer-2 input broadcast via wave32 shuffles;
// final dot reduced with shfl_xor tree; lane 0 writes tanh(.)
// ---------------------------------------------------------------------------
__global__ void nnue_value(const float* __restrict__ h,
                           const float* __restrict__ w1, const float* __restrict__ b1,
                           const float* __restrict__ w2, const float* __restrict__ b2,
                           const float* __restrict__ w3, const float* __restrict__ b3,
                           float* __restrict__ val) {
  const int wave = threadIdx.x >> 5;
  const int lane = threadIdx.x & 31;
  const int b = blockIdx.x * 8 + wave;

  // layer 1: v1[lane] = clip(dot(h[b], w1[lane]) + b1[lane], 0, 1)
  const float4* hv = (const float4*)(h  + (long long)b    * NNUE_L1);
  const float4* wv = (const float4*)(w1 + (long long)lane * NNUE_L1);
  float acc = b1[lane];
  #pragma unroll 8
  for (int t = 0; t < NNUE_L1 / 4; ++t) {
    const float4 hh = hv[t];
    const float4 ww = wv[t];
    acc = fmaf(hh.x, ww.x, acc);
    acc = fmaf(hh.y, ww.y, acc);
    acc = fmaf(hh.z, ww.z, acc);
    acc = fmaf(hh.w, ww.w, acc);
  }
  const float v1 = __builtin_fminf(__builtin_fmaxf(acc, 0.f), 1.f);

  // layer 2: v2[lane] = clip(dot(v1, w2[lane]) + b2[lane], 0, 1)
  float acc2 = b2[lane];
  const float* w2row = w2 + lane * 32;
  #pragma unroll
  for (int i = 0; i < 32; ++i) {
    const float vi = __shfl(v1, i, 32);
    acc2 = fmaf(vi, w2row[i], acc2);
  }
  const float v2c = __builtin_fminf(__builtin_fmaxf(acc2, 0.f), 1.f);

  // layer 3: val = tanh(dot(v2, w3) + b3)
  float p = v2c * w3[lane];
  #pragma unroll
  for (int off = 16; off >= 1; off >>= 1) {
    p += __shfl_xor(p, off, 32);
  }
  if (lane == 0) {
    val[b] = tanhf(p + b3[0]);
  }
}

// ---------------------------------------------------------------------------
extern "C" void kernel_launch(void* const* d_in, const int* in_sizes, int n_in,
                              void* d_out, int out_size, void* d_ws, size_t ws_size,
                              hipStream_t stream) {
  const int*   indices = (const int*)  d_in[0];
  const float* emb     = (const float*)d_in[1];
  const float* pw      = (const float*)d_in[2];
  const float* pb      = (const float*)d_in[3];
  const float* w1      = (const float*)d_in[4];
  const float* b1      = (const float*)d_in[5];
  const float* w2      = (const float*)d_in[6];
  const float* b2      = (const float*)d_in[7];
  const float* w3      = (const float*)d_in[8];
  const float* b3      = (const float*)d_in[9];

  float* pol = (float*)d_out;                                   // [16384,225]
  float* val = (float*)d_out + (long long)NNUE_BATCH * NNUE_POL; // [16384]
  float* h   = (float*)d_ws;                                    // 8 MB scratch

  // Phase 1: embedding-bag sum -> h
  nnue_embed_bag<<<NNUE_BATCH / 8, 256, 0, stream>>>(indices, emb, h);

  // Phase 2: policy GEMM (WMMA f32 16x16x4), 1024 x 15 tiles of 16x16
  dim3 gridB(NNUE_BATCH / 16, (NNUE_POL + 15) / 16);
  nnue_policy_wmma<<<gridB, 32, 0, stream>>>(h, pw, pb, pol);

  // Phase 3: fused value head
  nnue_value<<<NNUE_BATCH / 8, 256, 0, stream>>>(h, w1, b1, w2, b2, w3, b3, val);
}